// BilinearPooling_53163105190129
// MI455X (gfx1250) — compile-verified
//
#include <hip/hip_runtime.h>
#include <hip/hip_bf16.h>

// Problem constants from the reference: B=32, C=256, H=W=64.
#define BB   32
#define CC   256
#define HH   64
#define WW   64
#define HW   (HH * WW)          // 4096
#define PSPLIT 8                // p-range split per (b, ctile) for occupancy

typedef __attribute__((ext_vector_type(2))) float v2f;
typedef __attribute__((ext_vector_type(8))) float v8f;

// ---------------------------------------------------------------------------
// Pass 1: build the scatter weight field A[b, p] (p = h*W + w), pre-scaled by
// 1/(H*W), and zero the output accumulator.
//
// Bilinear sampling + mean pooling is linear in `data`, so
//   out[b,c] = sum_p A[b,p] * data[b,c,p]
// where A[b,p] accumulates the bilinear weight every destination pixel places
// on source pixel p.  One block per batch; 4096 pixels scattered via LDS f32
// atomics (ds_add_f32).
// ---------------------------------------------------------------------------
__global__ __launch_bounds__(1024)
void bp_scatter_weights(const float* __restrict__ offset,   // [B,2,H,W]
                        const float* __restrict__ trans_std, // scalar
                        float* __restrict__ wfield,          // [B,HW] (ws)
                        float* __restrict__ out)             // [B,C] zeroed here
{
    __shared__ float sw[HW];
    const int b   = blockIdx.x;
    const int tid = threadIdx.x;

    for (int p = tid; p < HW; p += 1024) sw[p] = 0.0f;
    if (tid < CC) out[b * CC + tid] = 0.0f;
    __syncthreads();

    const float ts = fminf(fmaxf(trans_std[0], 0.001f), 0.01f);
    const float* offy = offset + (size_t)b * 2 * HW;
    const float* offx = offy + HW;

    for (int p = tid; p < HW; p += 1024) {
        const int i = p >> 6;        // h
        const int j = p & (WW - 1);  // w
        float y = (float)i + ts * offy[p] * (float)HH;
        float x = (float)j + ts * offx[p] * (float)WW;
        y = fminf(fmaxf(y, 0.0f), (float)(HH - 1));
        x = fminf(fmaxf(x, 0.0f), (float)(WW - 1));
        int y0 = min(max((int)floorf(y), 0), HH - 2);
        int x0 = min(max((int)floorf(x), 0), WW - 2);
        const float wy = y - (float)y0;
        const float wx = x - (float)x0;
        const int p00 = y0 * WW + x0;
        atomicAdd(&sw[p00],          (1.0f - wy) * (1.0f - wx));
        atomicAdd(&sw[p00 + 1],      (1.0f - wy) * wx);
        atomicAdd(&sw[p00 + WW],     wy * (1.0f - wx));
        atomicAdd(&sw[p00 + WW + 1], wy * wx);
    }
    __syncthreads();

    const float inv = 1.0f / (float)HW;   // fold mean-pool scale into weights
    for (int p = tid; p < HW; p += 1024)
        wfield[(size_t)b * HW + p] = sw[p] * inv;
}

// ---------------------------------------------------------------------------
// Pass 2: out[b, c0..c0+15] += sum_p A[b,p] * data[b,c,p] via
// V_WMMA_F32_16X16X4_F32 (f32 in, f32 acc — bit-compatible with scalar FMA).
//
// A-matrix (16x4, M=channel tile row, K=p offset), ISA VGPR layout:
//   lanes 0-15 : {K=0, K=1}  -> load data[c, p+0 .. p+1]   (b64 per lane)
//   lanes 16-31: {K=2, K=3}  -> load data[c, p+2 .. p+3]
// B-matrix (4x16) carries the weight w[p+k] broadcast across all 16 columns,
// filled with the mirrored half-wave layout (same address per half-wave ->
// single coalesced fetch).  Every column of D is then identical; lanes 0 and
// 16 hold the 16 channel partials and atomically accumulate into out.
//
// One wave handles one (b, ctile, p-chunk of 512) => 128 WMMAs streaming
// 16ch x 512p x 4B = 32 KB of data, fully linear.
// ---------------------------------------------------------------------------
__global__ __launch_bounds__(256)
void bp_wmma_reduce(const float* __restrict__ data,    // [B,C,HW]
                    const float* __restrict__ wfield,  // [B,HW]
                    float* __restrict__ out)           // [B,C]
{
    const int wave = (int)((blockIdx.x * blockDim.x + threadIdx.x) >> 5);
    const int lane = (int)(threadIdx.x & 31);

    // wave -> (b, ctile, pchunk):  32 * 16 * 8 = 4096 wave-tasks
    const int pc    = wave & (PSPLIT - 1);
    const int ctile = (wave / PSPLIT) & 15;
    const int b     = wave / (PSPLIT * 16);

    const int c0     = ctile * 16;
    const int pstart = pc * (HW / PSPLIT);
    const int pend   = pstart + (HW / PSPLIT);

    const int row  = c0 + (lane & 15);
    const int koff = (lane >> 4) << 1;            // 0 for lanes 0-15, 2 for 16-31

    const float* aptr = data   + ((size_t)(b * CC + row)) * HW;
    const float* bptr = wfield + (size_t)b * HW;

    v8f acc = {0.f, 0.f, 0.f, 0.f, 0.f, 0.f, 0.f, 0.f};

    for (int p = pstart; p < pend; p += 16) {
        // hint the streaming read-ahead on the heavy operand
        __builtin_prefetch(aptr + p + 512, 0, 0);
#pragma unroll
        for (int q = 0; q < 16; q += 4) {
            const v2f a = *(const v2f*)(aptr + p + q + koff);
            const v2f w = *(const v2f*)(bptr + p + q + koff);
            acc = __builtin_amdgcn_wmma_f32_16x16x4_f32(
                /*neg_a=*/false, a, /*neg_b=*/false, w,
                /*c_mod=*/(short)0, acc, /*reuse_a=*/false, /*reuse_b=*/false);
        }
    }

    // All D columns are identical; lane 0 carries rows c0..c0+7 (VGPR j -> M=j),
    // lane 16 carries rows c0+8..c0+15.
    if ((lane & 15) == 0) {
        float* o = out + (size_t)b * CC + c0 + ((lane >> 4) << 3);
#pragma unroll
        for (int j = 0; j < 8; ++j)
            atomicAdd(&o[j], acc[j]);
    }
}

extern "C" void kernel_launch(void* const* d_in, const int* in_sizes, int n_in,
                              void* d_out, int out_size, void* d_ws, size_t ws_size,
                              hipStream_t stream) {
    const float* data      = (const float*)d_in[0];  // [32,256,64,64]
    const float* offset    = (const float*)d_in[1];  // [32,2,64,64]
    const float* trans_std = (const float*)d_in[2];  // scalar
    float* out    = (float*)d_out;                   // [32,256]
    float* wfield = (float*)d_ws;                    // needs 32*4096*4 = 512 KB

    (void)in_sizes; (void)n_in; (void)out_size; (void)ws_size;

    // Pass 1: weight scatter (one block per batch) + zero the output.
    bp_scatter_weights<<<BB, 1024, 0, stream>>>(offset, trans_std, wfield, out);

    // Pass 2: WMMA weighted reduction.
    // 32 b * 16 ctiles * 8 psplits = 4096 waves = 512 blocks of 8 waves.
    const int total_waves = BB * (CC / 16) * PSPLIT;
    bp_wmma_reduce<<<total_waves / 8, 256, 0, stream>>>(data, wfield, out);
}